// DNAConvEncoder_4801773437673
// MI455X (gfx1250) — compile-verified
//
#include <hip/hip_runtime.h>

typedef float v2f __attribute__((ext_vector_type(2)));
typedef float v8f __attribute__((ext_vector_type(8)));

// ---- degree: deg[dst[e]] += 1.0 (hardware global_atomic_add_f32) ----
__global__ void tag_deg(const int* __restrict__ dst, float* __restrict__ deg, int E) {
    int e = blockIdx.x * blockDim.x + threadIdx.x;
    if (e < E) unsafeAtomicAdd(&deg[dst[e]], 1.0f);
}

// ---- dis[i] = deg>0 ? rsqrt(max(deg,1)) : 0 ----
__global__ void tag_dis(const float* __restrict__ deg, float* __restrict__ dis, int N) {
    int i = blockIdx.x * blockDim.x + threadIdx.x;
    if (i < N) {
        float d = deg[i];
        dis[i] = (d > 0.0f) ? rsqrtf(fmaxf(d, 1.0f)) : 0.0f;
    }
}

// ---- norm[e] = dis[src[e]] * dis[dst[e]] ----
__global__ void tag_norm(const int* __restrict__ src, const int* __restrict__ dst,
                         const float* __restrict__ dis, float* __restrict__ nrm, int E) {
    int e = blockIdx.x * blockDim.x + threadIdx.x;
    if (e < E) nrm[e] = dis[src[e]] * dis[dst[e]];
}

// ---- one hop: out[dst[e], :] += h[src[e], :] * norm[e]  (D = 128)
// 32 lanes per edge, 4 floats each -> coalesced 512B gather + scatter per edge,
// buffers are L2-resident (51MB each vs 192MB L2).
__global__ void tag_prop(const float* __restrict__ h, const int* __restrict__ src,
                         const int* __restrict__ dst, const float* __restrict__ nrm,
                         float* __restrict__ out, int E) {
    long long t = (long long)blockIdx.x * blockDim.x + threadIdx.x;
    int e = (int)(t >> 5);
    if (e >= E) return;
    int d4 = ((int)t & 31) << 2;                 // feature offset 0..124
    int s  = src[e];
    int dd = dst[e];
    float w = nrm[e];
    const float4 v = *(const float4*)(h + (size_t)s * 128 + d4);
    float* o = out + (size_t)dd * 128 + d4;
    unsafeAtomicAdd(o + 0, v.x * w);
    unsafeAtomicAdd(o + 1, v.y * w);
    unsafeAtomicAdd(o + 2, v.z * w);
    unsafeAtomicAdd(o + 3, v.w * w);
}

// ---- C[16x16 tile] (+)= A[M x 128] @ W[128 x NOUT], fp32 WMMA ----
// One wave (EXEC all ones) per 16x16 output tile; K-loop in steps of 4 with
// v_wmma_f32_16x16x4_f32.
// A frag (32-bit 16x4 ISA layout): lanes 0-15 hold K=k0,k0+1; lanes 16-31 K=k0+2,k0+3.
// B frag mirrored over columns. C/D: 8 VGPRs, lanes 0-15 -> M=r, lanes 16-31 -> M=8+r.
// NOUT / ACC / BIAS / RELU are compile-time so all addresses fold to immediate
// offsets and the epilogue is branch-free straight-line code.
template <int NOUT, bool ACC, bool BIAS, bool RELU>
__global__ void tag_gemm(const float* __restrict__ A, const float* __restrict__ W,
                         float* __restrict__ C, const float* __restrict__ bias) {
    const int lane = threadIdx.x;                // 0..31, single wave
    const int tm = blockIdx.x, tn = blockIdx.y;
    const int lo = lane & 15;
    const int hi = lane >> 4;                    // 0 or 1
    const int arow = tm * 16 + lo;
    const int bcol = tn * 16 + lo;
    const int crow = tm * 16 + hi * 8;
    const int ccol = tn * 16 + lo;

    v8f c = {};
    if (ACC) {
#pragma unroll
        for (int r = 0; r < 8; ++r)
            c[r] = C[(size_t)(crow + r) * NOUT + ccol];
    }

    const float* arowp = A + (size_t)arow * 128 + hi * 2;
    const float* bp0   = W + (size_t)(hi * 2) * NOUT + bcol;
#pragma unroll 8
    for (int k0 = 0; k0 < 128; k0 += 4) {
        v2f a = *(const v2f*)(arowp + k0);       // contiguous float2 per lane
        v2f b;
        b.x = bp0[(size_t)k0 * NOUT];            // K = k0 + hi*2
        b.y = bp0[(size_t)k0 * NOUT + NOUT];     // K = k0 + hi*2 + 1
        c = __builtin_amdgcn_wmma_f32_16x16x4_f32(
                /*neg_a=*/false, a, /*neg_b=*/false, b,
                /*c_mod=*/(short)0, c, /*reuse_a=*/false, /*reuse_b=*/false);
    }

    const float bv = BIAS ? bias[ccol] : 0.0f;
#pragma unroll
    for (int r = 0; r < 8; ++r) {
        float v = c[r];
        if (BIAS) v += bv;
        if (RELU) v = fmaxf(v, 0.0f);
        C[(size_t)(crow + r) * NOUT + ccol] = v;
    }
}

extern "C" void kernel_launch(void* const* d_in, const int* in_sizes, int n_in,
                              void* d_out, int out_size, void* d_ws, size_t ws_size,
                              hipStream_t stream) {
    const float* x  = (const float*)d_in[0];
    const int*   ei = (const int*)d_in[1];
    const float* W1 = (const float*)d_in[2];
    const float* b1 = (const float*)d_in[3];
    const float* W2 = (const float*)d_in[4];
    const float* b2 = (const float*)d_in[5];

    const int N = in_sizes[0] / 128;   // 100000 (multiple of 16)
    const int E = in_sizes[1] / 2;     // 1600000
    const int* src = ei;               // edge_index row 0
    const int* dst = ei + E;           // edge_index row 1

    float* ws   = (float*)d_ws;
    float* deg  = ws;                              // N
    float* dis  = deg + N;                         // N
    float* nrm  = dis + N;                         // E
    float* bufA = nrm + E;                         // N*128
    float* bufB = bufA + (size_t)N * 128;          // N*128
    float* acc  = bufB + (size_t)N * 128;          // N*128
    float* out  = (float*)d_out;                   // N*16

    const int T = 256;
    const size_t featBytes = (size_t)N * 128 * sizeof(float);
    const long long pth = (long long)E * 32;
    const int pblk = (int)((pth + T - 1) / T);
    const dim3 g1(N / 16, 128 / 16);               // layer-1 GEMM tiles
    const dim3 g2(N / 16, 1);                      // layer-2 GEMM tiles (NOUT=16)

    // ---- normalization ----
    hipMemsetAsync(deg, 0, (size_t)N * sizeof(float), stream);
    tag_deg <<<(E + T - 1) / T, T, 0, stream>>>(dst, deg, E);
    tag_dis <<<(N + T - 1) / T, T, 0, stream>>>(deg, dis, N);
    tag_norm<<<(E + T - 1) / T, T, 0, stream>>>(src, dst, dis, nrm, E);

    // ---- layer 1: acc = sum_k (A^k x) @ W1[k]; h = relu(acc + b1) (in place) ----
    tag_gemm<128, false, false, false><<<g1, 32, 0, stream>>>(x,    W1 + 0 * 128 * 128, acc, nullptr);
    hipMemsetAsync(bufA, 0, featBytes, stream);
    tag_prop<<<pblk, T, 0, stream>>>(x, src, dst, nrm, bufA, E);
    tag_gemm<128, true,  false, false><<<g1, 32, 0, stream>>>(bufA, W1 + 1 * 128 * 128, acc, nullptr);
    hipMemsetAsync(bufB, 0, featBytes, stream);
    tag_prop<<<pblk, T, 0, stream>>>(bufA, src, dst, nrm, bufB, E);
    tag_gemm<128, true,  false, false><<<g1, 32, 0, stream>>>(bufB, W1 + 2 * 128 * 128, acc, nullptr);
    hipMemsetAsync(bufA, 0, featBytes, stream);
    tag_prop<<<pblk, T, 0, stream>>>(bufB, src, dst, nrm, bufA, E);
    tag_gemm<128, true,  true,  true ><<<g1, 32, 0, stream>>>(bufA, W1 + 3 * 128 * 128, acc, b1);

    // ---- layer 2: out = sum_k (A^k h) @ W2[k] + b2 ----
    tag_gemm<16, false, false, false><<<g2, 32, 0, stream>>>(acc,  W2 + 0 * 128 * 16, out, nullptr);
    hipMemsetAsync(bufA, 0, featBytes, stream);
    tag_prop<<<pblk, T, 0, stream>>>(acc, src, dst, nrm, bufA, E);
    tag_gemm<16, true,  false, false><<<g2, 32, 0, stream>>>(bufA, W2 + 1 * 128 * 16, out, nullptr);
    hipMemsetAsync(bufB, 0, featBytes, stream);
    tag_prop<<<pblk, T, 0, stream>>>(bufA, src, dst, nrm, bufB, E);
    tag_gemm<16, true,  false, false><<<g2, 32, 0, stream>>>(bufB, W2 + 2 * 128 * 16, out, nullptr);
    hipMemsetAsync(bufA, 0, featBytes, stream);
    tag_prop<<<pblk, T, 0, stream>>>(bufB, src, dst, nrm, bufA, E);
    tag_gemm<16, true,  true,  false><<<g2, 32, 0, stream>>>(bufA, W2 + 3 * 128 * 16, out, b2);

    (void)n_in; (void)out_size; (void)ws_size;
}